// MesoLayer_3521873183276
// MI455X (gfx1250) — compile-verified
//
#include <hip/hip_runtime.h>
#include <math.h>

// MI455X / gfx1250: wave32; WMMA f32 16x16x4 for the shared-weight GEMMs,
// async global->LDS (ASYNCcnt) double-buffered staging for the HBM stream.
typedef __attribute__((ext_vector_type(2))) float v2f;
typedef __attribute__((ext_vector_type(8))) float v8f;

#define B_TOTAL 32768
#define NNODE   78
#define EMB     30
#define PROJ    10
#define NSUB    7
#define WAVES_PER_BLOCK 2
#define NBLOCKS 1024

#define XT_STRIDE 34                       // even (8B rows) and conflict-free under lane-stride access
#define XT_FLOATS (16 * 25 * XT_STRIDE)    // 13600 (sized for L=25)
#define FS_OFFS   (2 * XT_FLOATS)          // 27200  fsum [b][g]  stride 34
#define SS_OFFS   (FS_OFFS + 16 * 34)      // 27744  S    [b][h]  stride 16
#define SV_OFFS   (SS_OFFS + 16 * 16)      // 28000  V    [b][g]  stride 34
#define LG_OFFS   (SV_OFFS + 16 * 34)      // 28544  att  [b][j]  stride 26
#define WAVE_FLOATS (LG_OFFS + 16 * 26)    // 28960
#define SMEM_FLOATS (384 + WAVES_PER_BLOCK * WAVE_FLOATS)

#define WAIT_ASYNC(n) asm volatile("s_wait_asynccnt %0" :: "i"(n) : "memory")

static __device__ __forceinline__ v8f wmma4(v2f a, v2f b, v8f c) {
    return __builtin_amdgcn_wmma_f32_16x16x4_f32(false, a, false, b, (short)0, c, false, false);
}

// Issue async 8B/lane copies of the (16 batch x L x 30) tile into LDS buffer.
// Every instance issues >= 38 instructions, which the pipelining sync relies on.
template<int L, int SOFF>
static __device__ __forceinline__ void stage_async(const float* __restrict__ x, int b0,
                                                   uint32_t xt_base_bytes, int lane) {
    constexpr int ROW2 = 15 * L;           // float2 per batch chunk (contiguous in memory)
    constexpr int TOT2 = 16 * ROW2;
    const uint64_t sbase = (uint64_t)(uintptr_t)(x + ((size_t)b0 * NNODE + SOFF) * EMB);
    for (int i = lane; i < TOT2; i += 32) {
        int b = i / ROW2; int r = i - b * ROW2;
        int j = r / 15;   int gg = r - j * 15;
        uint32_t lds  = xt_base_bytes + (uint32_t)(((j * 16 + b) * XT_STRIDE + 2 * gg) * 4);
        uint32_t voff = (uint32_t)(b * (NNODE * EMB * 4)) + (uint32_t)(r * 8);
        asm volatile("global_load_async_to_lds_b64 %0, %1, %2"
                     :: "v"(lds), "v"(voff), "s"(sbase) : "memory");
    }
}

template<int L, int SOFF, int SIDX>
static __device__ __forceinline__ void compute_sub(
        const float* __restrict__ XT, float* fs, float* sS, float* sV, float* lg,
        float* attPart, float* __restrict__ out,
        const v2f (&a1)[8], const v2f (&a2)[2][3],
        int b0, int lane, int hi2, int mrow) {

    // ---- fsum[b][g] = sum_j X[b,j,g] (float2 granularity) ----
    for (int i = lane; i < 240; i += 32) {
        int b = i / 15; int gg = i - b * 15;
        v2f a = {0.0f, 0.0f};
        for (int j = 0; j < L; ++j) {
            v2f xv = *(const v2f*)&XT[(j * 16 + b) * XT_STRIDE + 2 * gg];
            a.x += xv.x; a.y += xv.y;
        }
        *(v2f*)&fs[b * XT_STRIDE + 2 * gg] = a;
    }

    // ---- GEMM1 (WMMA): S(10x16) = W^T x Fsum(30x16), K padded to 32 ----
    v8f c1 = {0, 0, 0, 0, 0, 0, 0, 0};
#pragma unroll
    for (int kk = 0; kk < 8; ++kk) {
        int k0 = kk * 4 + hi2 * 2;
        v2f bf = *(const v2f*)&fs[mrow * XT_STRIDE + k0];   // in-row padding keeps this in-bounds
        if (k0 >= EMB) { bf.x = 0.0f; bf.y = 0.0f; }
        c1 = wmma4(a1[kk], bf, c1);
    }
#pragma unroll
    for (int r = 0; r < 8; ++r)                 // unconditional: sS padded to 16 slots
        sS[mrow * 16 + hi2 * 8 + r] = c1[r];

    // ---- GEMM2 (WMMA): V(30x16) = W x S(10x16), two M-tiles, K padded to 12 ----
#pragma unroll
    for (int mt = 0; mt < 2; ++mt) {
        v8f c2 = {0, 0, 0, 0, 0, 0, 0, 0};
#pragma unroll
        for (int kk = 0; kk < 3; ++kk) {
            int k0 = kk * 4 + hi2 * 2;
            v2f bf = *(const v2f*)&sS[mrow * 16 + k0];
            if (k0 >= PROJ) { bf.x = 0.0f; bf.y = 0.0f; }
            c2 = wmma4(a2[mt][kk], bf, c2);
        }
#pragma unroll
        for (int r = 0; r < 8; ++r)             // unconditional: sV rows hold 34 slots
            sV[mrow * XT_STRIDE + mt * 16 + hi2 * 8 + r] = c2[r];
    }

    // ---- logits[b][j] = X[b,j,:] . V[b,:] ----
    for (int i = lane; i < 16 * L; i += 32) {
        int b = i & 15; int j = i >> 4;         // lanes consecutive in b -> conflict-free
        const v2f* xr = (const v2f*)&XT[(j * 16 + b) * XT_STRIDE];
        const v2f* vr = (const v2f*)&sV[b * XT_STRIDE];
        float a = 0.0f;
#pragma unroll
        for (int g2 = 0; g2 < 15; ++g2) {
            v2f xv = xr[g2]; v2f vv = vr[g2];
            a = fmaf(xv.x, vv.x, a);
            a = fmaf(xv.y, vv.y, a);
        }
        lg[b * 26 + j] = a;
    }

    // ---- softmax over j per batch (lanes 0..15) ----
    if (lane < 16) {
        int b = lane;
        float m = -3.4e38f;
        for (int j = 0; j < L; ++j) m = fmaxf(m, lg[b * 26 + j]);
        float sum = 0.0f;
        for (int j = 0; j < L; ++j) {
            float e = expf(lg[b * 26 + j] - m);
            lg[b * 26 + j] = e;
            sum += e;
        }
        float inv = 1.0f / sum;
        for (int j = 0; j < L; ++j) lg[b * 26 + j] *= inv;
    }

    // ---- per-block partial of sum_b att[b][j] (for coordinates) ----
    if (lane < L) {
        float a = 0.0f;
        for (int b = 0; b < 16; ++b) a += lg[b * 26 + lane];
        atomicAdd(&attPart[SOFF + lane], a);
    }

    // ---- feat[b][g] = sum_j att[b][j] * X[b,j,g]; float2 coalesced stores ----
    for (int i = lane; i < 240; i += 32) {
        int b = i / 15; int gg = i - b * 15;
        v2f acc = {0.0f, 0.0f};
        for (int j = 0; j < L; ++j) {
            float wj = lg[b * 26 + j];
            v2f xv = *(const v2f*)&XT[(j * 16 + b) * XT_STRIDE + 2 * gg];
            acc.x = fmaf(wj, xv.x, acc.x);
            acc.y = fmaf(wj, xv.y, acc.y);
        }
        *(v2f*)&out[((size_t)(b0 + b) * NSUB + SIDX) * EMB + 2 * gg] = acc;
    }
}

__global__ void meso_main(const float* __restrict__ x, const float* __restrict__ w,
                          float* __restrict__ out, float* __restrict__ part) {
    extern __shared__ float smem[];
    const int tid  = threadIdx.x;
    const int wave = tid >> 5;
    const int lane = tid & 31;
    const int hi2  = lane >> 4;
    const int mrow = lane & 15;

    float* w_lds   = smem;          // 300 (pad 304)
    float* attPart = smem + 304;    // 78 (pad 80)
    float* wb  = smem + 384 + wave * WAVE_FLOATS;
    float* XT0 = wb;
    float* XT1 = wb + XT_FLOATS;
    float* fs  = wb + FS_OFFS;
    float* sS  = wb + SS_OFFS;
    float* sV  = wb + SV_OFFS;
    float* lg  = wb + LG_OFFS;

    for (int i = tid; i < 300; i += blockDim.x) w_lds[i] = w[i];
    for (int i = tid; i < 78;  i += blockDim.x) attPart[i] = 0.0f;
    __syncthreads();

    // WMMA A-fragments of the shared weight (subgraph-invariant, kept in VGPRs).
    v2f a1[8];
#pragma unroll
    for (int kk = 0; kk < 8; ++kk) {
        int k0 = kk * 4 + hi2 * 2;
        a1[kk].x = (k0     < EMB && mrow < PROJ) ? w_lds[k0 * PROJ + mrow]       : 0.0f;
        a1[kk].y = (k0 + 1 < EMB && mrow < PROJ) ? w_lds[(k0 + 1) * PROJ + mrow] : 0.0f;
    }
    v2f a2[2][3];
#pragma unroll
    for (int mt = 0; mt < 2; ++mt)
#pragma unroll
        for (int kk = 0; kk < 3; ++kk) {
            int k0 = kk * 4 + hi2 * 2;
            int g  = mt * 16 + mrow;
            a2[mt][kk].x = (g < EMB && k0     < PROJ) ? w_lds[g * PROJ + k0]     : 0.0f;
            a2[mt][kk].y = (g < EMB && k0 + 1 < PROJ) ? w_lds[g * PROJ + k0 + 1] : 0.0f;
        }

    const uint32_t xt0b = (uint32_t)(uintptr_t)XT0;
    const uint32_t xt1b = (uint32_t)(uintptr_t)XT1;
    const int b0 = (blockIdx.x * WAVES_PER_BLOCK + wave) * 16;

    // Software pipeline: stage s+1 async into the other buffer while computing s.
    // s_wait_asynccnt 31 + in-order async completion + every stage issuing >=38
    // instructions guarantees the current buffer is fully resident.
    stage_async<5, 0>(x, b0, xt0b, lane);
    stage_async<9, 5>(x, b0, xt1b, lane);
    WAIT_ASYNC(31);
    compute_sub<5, 0, 0>(XT0, fs, sS, sV, lg, attPart, out, a1, a2, b0, lane, hi2, mrow);
    stage_async<9, 14>(x, b0, xt0b, lane);
    WAIT_ASYNC(31);
    compute_sub<9, 5, 1>(XT1, fs, sS, sV, lg, attPart, out, a1, a2, b0, lane, hi2, mrow);
    stage_async<25, 23>(x, b0, xt1b, lane);
    WAIT_ASYNC(31);
    compute_sub<9, 14, 2>(XT0, fs, sS, sV, lg, attPart, out, a1, a2, b0, lane, hi2, mrow);
    stage_async<9, 48>(x, b0, xt0b, lane);
    WAIT_ASYNC(31);
    compute_sub<25, 23, 3>(XT1, fs, sS, sV, lg, attPart, out, a1, a2, b0, lane, hi2, mrow);
    stage_async<9, 57>(x, b0, xt1b, lane);
    WAIT_ASYNC(31);
    compute_sub<9, 48, 4>(XT0, fs, sS, sV, lg, attPart, out, a1, a2, b0, lane, hi2, mrow);
    stage_async<12, 66>(x, b0, xt0b, lane);
    WAIT_ASYNC(31);
    compute_sub<9, 57, 5>(XT1, fs, sS, sV, lg, attPart, out, a1, a2, b0, lane, hi2, mrow);
    WAIT_ASYNC(0);
    compute_sub<12, 66, 6>(XT0, fs, sS, sV, lg, attPart, out, a1, a2, b0, lane, hi2, mrow);

    __syncthreads();
    for (int i = tid; i < 78; i += blockDim.x)
        part[(size_t)blockIdx.x * 78 + i] = attPart[i];
}

__global__ void meso_coords(const float* __restrict__ coords, const float* __restrict__ part,
                            float* __restrict__ out) {
    const int LEN[NSUB] = {5, 9, 9, 25, 9, 9, 12};
    const int OFF[NSUB] = {0, 5, 14, 23, 48, 57, 66};
    __shared__ float asum[80];
    int tid = threadIdx.x;
    if (tid < 78) {
        float s = 0.0f;
        for (int blk = 0; blk < NBLOCKS; ++blk) s += part[(size_t)blk * 78 + tid];
        asum[tid] = s;
    }
    __syncthreads();
    if (tid < NSUB * 3) {
        int s = tid / 3, k = tid - 3 * s;
        float acc = 0.0f;
        for (int j = 0; j < LEN[s]; ++j)
            acc += asum[OFF[s] + j] * coords[(OFF[s] + j) * 3 + k];
        out[(size_t)B_TOTAL * NSUB * EMB + tid] = acc * (1.0f / (float)B_TOTAL);
    }
}

extern "C" void kernel_launch(void* const* d_in, const int* in_sizes, int n_in,
                              void* d_out, int out_size, void* d_ws, size_t ws_size,
                              hipStream_t stream) {
    const float* x      = (const float*)d_in[0];
    const float* coords = (const float*)d_in[1];
    const float* w      = (const float*)d_in[2];
    float* out  = (float*)d_out;
    float* part = (float*)d_ws;   // NBLOCKS * 78 deterministic partials

    const size_t smem_bytes = (size_t)SMEM_FLOATS * sizeof(float);
    (void)hipFuncSetAttribute((const void*)meso_main,
                              hipFuncAttributeMaxDynamicSharedMemorySize,
                              (int)smem_bytes);

    meso_main<<<NBLOCKS, 32 * WAVES_PER_BLOCK, smem_bytes, stream>>>(x, w, out, part);
    meso_coords<<<1, 128, 0, stream>>>(coords, part, out);
}